// _PlaneWaveFFT_43602507989876
// MI455X (gfx1250) — compile-verified
//
#include <hip/hip_runtime.h>
#include <math.h>

// ---------------------------------------------------------------------------
// PlaneWaveFFT for MI455X (gfx1250):
//   QR (CholeskyQR, WMMA f32) -> scatter -> 3x DFT-as-matmul (WMMA f32)
// v2: fused output tiles (A read once), 2-chain complex accumulate via
//     B-fragment negation, K-split Gram for parallelism, global prefetch.
// ---------------------------------------------------------------------------

typedef float v2f __attribute__((ext_vector_type(2)));
typedef float v8f __attribute__((ext_vector_type(8)));

#define NKPT   4
#define NG     12000
#define NBAND  64
#define NAX    48
#define GRID3  (48*48*48)      // 110592
#define NBATCH (NKPT*NBAND)    // 256
#define KCH    8               // K-chunks for the Gram stage
#define CHLEN  (NG/KCH)        // 1500 (multiple of 4)

// D = A(16x4,f32) * B(4x16,f32) + C(16x16,f32) on one wave32.
__device__ __forceinline__ v8f wmma4(v2f a, v2f b, v8f c) {
  return __builtin_amdgcn_wmma_f32_16x16x4_f32(
      /*neg_a=*/false, a, /*neg_b=*/false, b,
      /*c_mod=*/(short)0, c, /*reuse_a=*/false, /*reuse_b=*/false);
}

// ---------------------------------------------------------------------------
// Twiddle matrix F[p][a] = exp(+2*pi*i * p*a / 48)
// ---------------------------------------------------------------------------
__global__ void k_twiddle(float2* __restrict__ F) {
  int t = blockIdx.x * blockDim.x + threadIdx.x;
  if (t >= NAX * NAX) return;
  int p = t / NAX, a = t % NAX;
  float ang = 6.28318530717958647692f * (float)((p * a) % NAX) / (float)NAX;
  F[t] = make_float2(cosf(ang), sinf(ang));
}

// ---------------------------------------------------------------------------
// idx[] = flattened positions where mask != 0, ascending (C) order.
// ---------------------------------------------------------------------------
__global__ void k_index(const unsigned char* __restrict__ mask,
                        int* __restrict__ idx) {
  __shared__ int cnt[256];
  const int t  = threadIdx.x;
  const int CH = (GRID3 + 255) / 256;            // 432
  const int lo = t * CH;
  const int hi = (lo + CH < GRID3) ? lo + CH : GRID3;
  int c = 0;
  for (int i = lo; i < hi; ++i) c += mask[i] ? 1 : 0;
  cnt[t] = c;
  __syncthreads();
  if (t == 0) {
    int s = 0;
    for (int i = 0; i < 256; ++i) { int v = cnt[i]; cnt[i] = s; s += v; }
  }
  __syncthreads();
  int off = cnt[t];
  for (int i = lo; i < hi; ++i)
    if (mask[i]) idx[off++] = i;
}

// ---------------------------------------------------------------------------
// Gram partials: Gp[ch][k] += W[k][chunk]^H * W[k][chunk]
// One wave per (chunk, k, ci row-of-tiles): A panel read once, 4 cj tiles.
// G_re = rr + ii (no negation), G_im = ar*bi + ai*(-br).
// ---------------------------------------------------------------------------
__global__ void k_gram(const float* __restrict__ wre,
                       const float* __restrict__ wim,
                       float2* __restrict__ Gp) {
  const int b  = blockIdx.x;                     // KCH * NKPT * 4 = 128 blocks
  const int ch = b >> 4;
  const int k  = (b >> 2) & 3;
  const int ci = (b & 3) * 16;
  const int L  = threadIdx.x;
  const int m  = L & 15;
  const int kboff = (L < 16) ? 0 : 2;
  const float* wrk = wre + (size_t)k * NG * NBAND;
  const float* wik = wim + (size_t)k * NG * NBAND;
  const int gbeg = ch * CHLEN, gend = gbeg + CHLEN;

  v8f accRe[4] = {{}, {}, {}, {}};
  v8f accIm[4] = {{}, {}, {}, {}};
  for (int g0 = gbeg; g0 < gend; g0 += 4) {
    const size_t r0 = (size_t)(g0 + kboff) * NBAND;
    const size_t r1 = r0 + NBAND;
    // A[m][k'] = W[g0+k'][ci+m]  (conjugated operand of the Gram product)
    v2f are = { wrk[r0 + ci + m], wrk[r1 + ci + m] };
    v2f aim = { wik[r0 + ci + m], wik[r1 + ci + m] };
#pragma unroll
    for (int cj = 0; cj < 4; ++cj) {
      const int c = cj * 16 + m;
      v2f bre  = { wrk[r0 + c], wrk[r1 + c] };
      v2f bim  = { wik[r0 + c], wik[r1 + c] };
      v2f bnre = { -bre[0], -bre[1] };
      accRe[cj] = wmma4(are, bre,  accRe[cj]);   // + ar*br
      accRe[cj] = wmma4(aim, bim,  accRe[cj]);   // + ai*bi
      accIm[cj] = wmma4(are, bim,  accIm[cj]);   // + ar*bi
      accIm[cj] = wmma4(aim, bnre, accIm[cj]);   // - ai*br
    }
  }
  const int n = L & 15, half = L >> 4;
  float2* Gk = Gp + ((size_t)ch * NKPT + k) * 64 * 64;
#pragma unroll
  for (int cj = 0; cj < 4; ++cj)
#pragma unroll
    for (int v = 0; v < 8; ++v) {
      const int row = ci + v + 8 * half;
      Gk[(size_t)row * 64 + cj * 16 + n] = make_float2(accRe[cj][v], accIm[cj][v]);
    }
}

// Sum the KCH Gram partials.
__global__ void k_greduce(const float2* __restrict__ Gp, float2* __restrict__ G) {
  int t = blockIdx.x * blockDim.x + threadIdx.x;
  if (t >= NKPT * 64 * 64) return;
  float2 s = make_float2(0.f, 0.f);
  for (int ch = 0; ch < KCH; ++ch) {
    float2 v = Gp[(size_t)ch * NKPT * 64 * 64 + t];
    s.x += v.x; s.y += v.y;
  }
  G[t] = s;
}

// ---------------------------------------------------------------------------
// Per-k: Cholesky G = L L^H, invert L, output U = R^{-1} = (L^{-1})^H.
// ---------------------------------------------------------------------------
__global__ void k_chol(const float2* __restrict__ G, float2* __restrict__ U) {
  __shared__ float2 Lm[64][64];
  __shared__ float2 Li[64][64];
  __shared__ float2 scol[64];
  const int k = blockIdx.x, t = threadIdx.x;
  const float2* Gk = G + (size_t)k * 64 * 64;
  for (int j = 0; j < 64; ++j) Lm[t][j] = Gk[t * 64 + j];
  __syncthreads();

  for (int j = 0; j < 64; ++j) {
    float2 s = make_float2(0.f, 0.f);
    if (t >= j) {
      s = Lm[t][j];
      for (int p = 0; p < j; ++p) {              // s -= L[t][p] * conj(L[j][p])
        float2 a = Lm[t][p], c = Lm[j][p];
        s.x -= a.x * c.x + a.y * c.y;
        s.y -= a.y * c.x - a.x * c.y;
      }
    }
    scol[t] = s;
    __syncthreads();
    const float d = sqrtf(fmaxf(scol[j].x, 1e-30f));
    if (t == j)      Lm[j][j] = make_float2(d, 0.f);
    else if (t > j)  Lm[t][j] = make_float2(scol[t].x / d, scol[t].y / d);
    __syncthreads();
  }

  {                                              // thread t solves L x = e_t
    const int j = t;
    for (int i = 0; i < 64; ++i) Li[i][j] = make_float2(0.f, 0.f);
    Li[j][j] = make_float2(1.0f / Lm[j][j].x, 0.f);
    for (int i = j + 1; i < 64; ++i) {
      float2 s = make_float2(0.f, 0.f);
      for (int p = j; p < i; ++p) {
        float2 a = Lm[i][p], x = Li[p][j];
        s.x += a.x * x.x - a.y * x.y;
        s.y += a.x * x.y + a.y * x.x;
      }
      const float di = Lm[i][i].x;
      Li[i][j] = make_float2(-s.x / di, -s.y / di);
    }
  }
  __syncthreads();

  float2* Uk = U + (size_t)k * 64 * 64;          // U[p][q] = conj(Li[q][p])
  for (int q = 0; q < 64; ++q) {
    float2 v = (q >= t) ? Li[q][t] : make_float2(0.f, 0.f);
    Uk[t * 64 + q] = make_float2(v.x, -v.y);
  }
}

// ---------------------------------------------------------------------------
// Q = W * U (12000x64 @ 64x64 complex per k), fused over all 4 column tiles,
// scattered straight from the D-fragments into the zeroed dense grid.
// re = ar*br + ai*(-bi),  im = ar*bi + ai*br  (2 chains per tile).
// ---------------------------------------------------------------------------
__global__ void k_qscatter(const float* __restrict__ wre,
                           const float* __restrict__ wim,
                           const float2* __restrict__ U,
                           const int* __restrict__ idx,
                           float2* __restrict__ gridbuf) {
  const int b  = blockIdx.x;                     // NKPT * 750 = 3000 blocks
  const int k  = b / 750;
  const int g0 = (b % 750) * 16;
  const int L  = threadIdx.x;
  const int m  = L & 15;
  const int kboff = (L < 16) ? 0 : 2;
  const float* wrk = wre + (size_t)k * NG * NBAND + (size_t)(g0 + m) * NBAND;
  const float* wik = wim + (size_t)k * NG * NBAND + (size_t)(g0 + m) * NBAND;
  const float2* Uk = U + (size_t)k * 64 * 64;

  v8f accRe[4] = {{}, {}, {}, {}};
  v8f accIm[4] = {{}, {}, {}, {}};
#pragma unroll 4
  for (int p0 = 0; p0 < 64; p0 += 4) {
    const int kk = p0 + kboff;
    v2f are = { wrk[kk], wrk[kk + 1] };          // contiguous pair (b64 load)
    v2f aim = { wik[kk], wik[kk + 1] };
#pragma unroll
    for (int cj = 0; cj < 4; ++cj) {
      float2 b0 = Uk[(size_t)kk * 64 + cj * 16 + m];
      float2 b1 = Uk[(size_t)(kk + 1) * 64 + cj * 16 + m];
      v2f bre  = { b0.x, b1.x };
      v2f bim  = { b0.y, b1.y };
      v2f bnim = { -b0.y, -b1.y };
      accRe[cj] = wmma4(are, bre,  accRe[cj]);
      accRe[cj] = wmma4(aim, bnim, accRe[cj]);
      accIm[cj] = wmma4(are, bim,  accIm[cj]);
      accIm[cj] = wmma4(aim, bre,  accIm[cj]);
    }
  }
  const int n = L & 15, half = L >> 4;
#pragma unroll
  for (int v = 0; v < 8; ++v) {
    const int g   = g0 + v + 8 * half;
    const int pos = idx[g];
#pragma unroll
    for (int cj = 0; cj < 4; ++cj)
      gridbuf[((size_t)(k * NBAND + cj * 16 + n)) * GRID3 + pos] =
          make_float2(accRe[cj][v], accIm[cj][v]);
  }
}

// ---------------------------------------------------------------------------
// One DFT pass along the axis of compile-time stride S (elements):
//   out[row, a] = sum_p in[row, p] * F[p][a]   (complex 48-point DFT)
// Fully fused: one wave computes all 48 outputs of its 16 rows (A read once).
// flat addr of (row, p) = base(row) + p*S, base(row) = (row/S)*(48*S)+row%S.
// ---------------------------------------------------------------------------
template <int S>
__global__ void k_dft(const float2* __restrict__ in,
                      float2* __restrict__ out,
                      const float2* __restrict__ F) {
  const int row0 = blockIdx.x * 16;              // R/16 = 36864 blocks
  const int L    = threadIdx.x;
  const int m    = L & 15;
  const int kboff = (L < 16) ? 0 : 2;

  const int rowm = row0 + m;
  const size_t baseA = (size_t)(rowm / S) * (size_t)(NAX * S) + (size_t)(rowm % S);

  // Pull the tail of this pencil toward the caches while WMMAs start.
  __builtin_prefetch((const void*)(in + baseA + (size_t)16 * S), 0, 1);
  __builtin_prefetch((const void*)(in + baseA + (size_t)32 * S), 0, 1);

  v8f accRe[3] = {{}, {}, {}};
  v8f accIm[3] = {{}, {}, {}};
#pragma unroll 4
  for (int p0 = 0; p0 < NAX; p0 += 4) {
    const int kk = p0 + kboff;
    float2 a0 = in[baseA + (size_t)kk * S];
    float2 a1 = in[baseA + (size_t)(kk + 1) * S];
    v2f are = { a0.x, a1.x }, aim = { a0.y, a1.y };
#pragma unroll
    for (int t = 0; t < 3; ++t) {
      float2 b0 = F[kk * NAX + t * 16 + m];
      float2 b1 = F[(kk + 1) * NAX + t * 16 + m];
      v2f bre  = { b0.x, b1.x };
      v2f bim  = { b0.y, b1.y };
      v2f bnim = { -b0.y, -b1.y };
      accRe[t] = wmma4(are, bre,  accRe[t]);
      accRe[t] = wmma4(aim, bnim, accRe[t]);
      accIm[t] = wmma4(are, bim,  accIm[t]);
      accIm[t] = wmma4(aim, bre,  accIm[t]);
    }
  }
  const int n = L & 15, half = L >> 4;
  size_t baseD[8];
#pragma unroll
  for (int v = 0; v < 8; ++v) {
    const int row = row0 + v + 8 * half;
    baseD[v] = (size_t)(row / S) * (size_t)(NAX * S) + (size_t)(row % S);
  }
#pragma unroll
  for (int t = 0; t < 3; ++t)
#pragma unroll
    for (int v = 0; v < 8; ++v)
      out[baseD[v] + (size_t)(t * 16 + n) * S] =
          make_float2(accRe[t][v], accIm[t][v]);
}

// ---------------------------------------------------------------------------
extern "C" void kernel_launch(void* const* d_in, const int* in_sizes, int n_in,
                              void* d_out, int out_size, void* d_ws, size_t ws_size,
                              hipStream_t stream) {
  const float*         wre  = (const float*)d_in[0];          // (1,4,12000,64)
  const float*         wim  = (const float*)d_in[1];
  const unsigned char* mask = (const unsigned char*)d_in[2];  // bool (48,48,48)
  // d_in[3] = k_vector_grid: only its last-dim length (=3) matters; unused.
  (void)in_sizes; (void)n_in; (void)out_size; (void)ws_size;

  char* ws = (char*)d_ws;
  size_t off = 0;
  auto wsalloc = [&](size_t bytes) -> void* {
    void* p = ws + off;
    off = (off + bytes + 255) & ~(size_t)255;
    return p;
  };
  float2* gridbuf = (float2*)wsalloc((size_t)NBATCH * GRID3 * sizeof(float2)); // 226.5 MB
  int*    idx     = (int*)   wsalloc((size_t)NG * sizeof(int));
  float2* Gp      = (float2*)wsalloc((size_t)KCH * NKPT * 64 * 64 * sizeof(float2));
  float2* G       = (float2*)wsalloc((size_t)NKPT * 64 * 64 * sizeof(float2));
  float2* U       = (float2*)wsalloc((size_t)NKPT * 64 * 64 * sizeof(float2));
  float2* F       = (float2*)wsalloc((size_t)NAX * NAX * sizeof(float2));
  float2* outc    = (float2*)d_out;                           // interleaved complex64

  k_twiddle<<<(NAX * NAX + 255) / 256, 256, 0, stream>>>(F);
  k_index<<<1, 256, 0, stream>>>(mask, idx);
  k_gram<<<KCH * NKPT * 4, 32, 0, stream>>>(wre, wim, Gp);
  k_greduce<<<(NKPT * 64 * 64 + 255) / 256, 256, 0, stream>>>(Gp, G);
  k_chol<<<NKPT, 64, 0, stream>>>(G, U);
  hipMemsetAsync(gridbuf, 0, (size_t)NBATCH * GRID3 * sizeof(float2), stream);
  k_qscatter<<<NKPT * 750, 32, 0, stream>>>(wre, wim, U, idx, gridbuf);

  const int R = NBATCH * GRID3 / NAX;            // 589824 lines per pass
  k_dft<1>   <<<R / 16, 32, 0, stream>>>(gridbuf, outc,    F);  // x axis
  k_dft<48>  <<<R / 16, 32, 0, stream>>>(outc,    gridbuf, F);  // y axis
  k_dft<2304><<<R / 16, 32, 0, stream>>>(gridbuf, outc,    F);  // z axis
}